// TopKPool3D_31482110280280
// MI455X (gfx1250) — compile-verified
//
#include <hip/hip_runtime.h>
#include <hip/hip_bf16.h>

// Problem constants (from reference): B=4, C=256, D=32, H=64, W=64
#define BATCH   4
#define C_DIM   256
#define N_VOX   (32 * 64 * 64)   // 131072 voxels per batch
#define TOPK    256

typedef __attribute__((ext_vector_type(2))) float v2f;
typedef __attribute__((ext_vector_type(8))) float v8f;

// ---------------------------------------------------------------------------
// Phase 1: per-voxel score s[b,n] = sum_c Fmap[b,c,n] * w[c]
// One wave computes scores for 16 consecutive voxels using
// V_WMMA_F32_16X16X4_F32 accumulating over the channel (K) dimension.
//
// A (16x4, M=voxel, K=channel): lanes 0-15 hold M=lane with K=0 (x), K=1 (y);
// lanes 16-31 hold M=lane-16 with K=2 (x), K=3 (y)   [ISA 7.12.2].
// B (4x16): w[c+k] broadcast across all 16 columns, mirrored K split per
// half-wave -> every column of D equals the 16 voxel scores.
// D (16x16 f32, 8 VGPRs): column 0 lives in lane 0 (M=0..7 across acc[0..7])
// and lane 16 (M=8..15).
// ---------------------------------------------------------------------------
__global__ void __launch_bounds__(256)
topk_score_wmma(const float* __restrict__ Fmap,
                const float* __restrict__ w,
                float* __restrict__ scores) {
    const int lane     = threadIdx.x & 31;
    const int waveInBl = threadIdx.x >> 5;
    const int waveGlb  = blockIdx.x * (blockDim.x >> 5) + waveInBl;

    const int wavesPerBatch = N_VOX / 16;            // 8192
    const int b  = waveGlb / wavesPerBatch;
    const int t  = waveGlb % wavesPerBatch;
    const long n0 = (long)t * 16;

    const float* base = Fmap + (long)b * C_DIM * N_VOX;

    const int m  = lane & 15;          // voxel row within tile
    const int kb = (lane >> 4) * 2;    // K base: 0 for lanes 0-15, 2 for 16-31

    v8f acc = {};
    for (int c = 0; c < C_DIM; c += 4) {
        v2f a, bm;
        a.x  = base[(long)(c + kb)     * N_VOX + n0 + m];
        a.y  = base[(long)(c + kb + 1) * N_VOX + n0 + m];
        bm.x = w[c + kb];
        bm.y = w[c + kb + 1];
        acc = __builtin_amdgcn_wmma_f32_16x16x4_f32(
                  /*neg_a=*/false, a, /*neg_b=*/false, bm,
                  /*c_mod=*/(short)0, acc, /*reuse_a=*/false, /*reuse_b=*/false);
    }

    // Write column 0 of D: 16 voxel scores. (Bias is rank-invariant: omitted.)
    float* sout = scores + (long)b * N_VOX + n0;
    if (lane == 0) {
#pragma unroll
        for (int r = 0; r < 8; ++r) sout[r] = acc[r];
    } else if (lane == 16) {
#pragma unroll
        for (int r = 0; r < 8; ++r) sout[8 + r] = acc[r];
    }
}

// ---------------------------------------------------------------------------
// Phase 2: per-batch top-k (k=256) via 4-pass 8-bit radix select.
// Map f32 -> order-preserving u32, find the exact k-th largest bit pattern,
// then emit all strictly-greater indices plus the needed number of equals.
// One block per batch.
// ---------------------------------------------------------------------------
__device__ __forceinline__ unsigned f2ord(float f) {
    unsigned u = __float_as_uint(f);
    return (u & 0x80000000u) ? ~u : (u | 0x80000000u);
}

__global__ void __launch_bounds__(1024)
topk_radix_select(const float* __restrict__ scores, int* __restrict__ idx_out) {
    const int b = blockIdx.x;
    const float* s = scores + (long)b * N_VOX;
    const int tid = threadIdx.x;
    const int nth = blockDim.x;

    __shared__ unsigned hist[256];
    __shared__ unsigned prefix_s;
    __shared__ int      kRem_s;
    __shared__ int      cntG_s, cntE_s;

    if (tid == 0) { prefix_s = 0u; kRem_s = TOPK; }
    __syncthreads();

    for (int shift = 24; shift >= 0; shift -= 8) {
        for (int i = tid; i < 256; i += nth) hist[i] = 0u;
        __syncthreads();

        const unsigned prefix = prefix_s;
        const unsigned mask   = (shift == 24) ? 0u : (0xFFFFFFFFu << (shift + 8));
        for (int i = tid; i < N_VOX; i += nth) {
            unsigned u = f2ord(s[i]);
            if ((u & mask) == prefix)
                atomicAdd(&hist[(u >> shift) & 255u], 1u);
        }
        __syncthreads();

        if (tid == 0) {
            int kRem = kRem_s;
            unsigned cum = 0;
            int bin = 255;
            for (; bin >= 0; --bin) {
                unsigned h = hist[bin];
                if (cum + h >= (unsigned)kRem) break;
                cum += h;
            }
            if (bin < 0) bin = 0;                       // defensive
            prefix_s = prefix | ((unsigned)bin << shift);
            kRem_s   = kRem - (int)cum;
        }
        __syncthreads();
    }

    const unsigned uT    = prefix_s;   // bit pattern of k-th largest score
    const int     needEq = kRem_s;     // equals still required
    if (tid == 0) { cntG_s = 0; cntE_s = 0; }
    __syncthreads();

    int* dst = idx_out + b * TOPK;
    for (int i = tid; i < N_VOX; i += nth) {
        unsigned u = f2ord(s[i]);
        if (u > uT) {
            int p = atomicAdd(&cntG_s, 1);
            if (p < TOPK - needEq) dst[p] = i;
        } else if (u == uT) {
            int p = atomicAdd(&cntE_s, 1);
            if (p < needEq) dst[(TOPK - needEq) + p] = i;
        }
    }
}

// ---------------------------------------------------------------------------
// Phase 3: bitonic-sort the 256 indices (fixed FP summation order =>
// deterministic output), then out[b,c] = mean over top-k of Fmap[b,c,idx].
// One block per batch, 256 threads (= C).
// ---------------------------------------------------------------------------
__global__ void __launch_bounds__(256)
topk_gather_mean(const float* __restrict__ Fmap,
                 const int* __restrict__ idx,
                 float* __restrict__ out) {
    const int b = blockIdx.x;
    const int c = threadIdx.x;

    __shared__ int sidx[TOPK];
    {
        int v = idx[b * TOPK + c];
        if ((unsigned)v >= (unsigned)N_VOX) v = 0;   // defensive clamp
        sidx[c] = v;
    }
    __syncthreads();

    // Bitonic sort ascending (256 elements, 256 threads)
    for (int size = 2; size <= TOPK; size <<= 1) {
        for (int stride = size >> 1; stride > 0; stride >>= 1) {
            int i = c, j = i ^ stride;
            if (j > i) {
                bool up = ((i & size) == 0);
                int a = sidx[i], d = sidx[j];
                if ((a > d) == up) { sidx[i] = d; sidx[j] = a; }
            }
            __syncthreads();
        }
    }

    const float* base = Fmap + (long)b * C_DIM * N_VOX + (long)c * N_VOX;
    float sum = 0.0f;
#pragma unroll 4
    for (int j = 0; j < TOPK; ++j) sum += base[sidx[j]];
    out[b * C_DIM + c] = sum * (1.0f / (float)TOPK);
}

// ---------------------------------------------------------------------------
// Launch. Workspace layout: [0, 2MB) f32 scores, [2MB, 2MB+4KB) int indices.
// ---------------------------------------------------------------------------
extern "C" void kernel_launch(void* const* d_in, const int* in_sizes, int n_in,
                              void* d_out, int out_size, void* d_ws, size_t ws_size,
                              hipStream_t stream) {
    const float* Fmap = (const float*)d_in[0];
    const float* w    = (const float*)d_in[1];
    // d_in[2] = score_b: constant bias, rank-invariant, never reaches output.

    float* scores = (float*)d_ws;
    int*   idxbuf = (int*)((char*)d_ws + (size_t)BATCH * N_VOX * sizeof(float));
    float* out    = (float*)d_out;

    // Phase 1: 4*131072/16 = 32768 waves, 8 waves/block -> 4096 blocks.
    const int wavesTotal = (BATCH * N_VOX) / 16;
    dim3 g1(wavesTotal / 8), b1(256);
    topk_score_wmma<<<g1, b1, 0, stream>>>(Fmap, w, scores);

    // Phase 2: one block per batch.
    topk_radix_select<<<BATCH, 1024, 0, stream>>>(scores, idxbuf);

    // Phase 3: one block per batch, 256 threads = C.
    topk_gather_mean<<<BATCH, 256, 0, stream>>>(Fmap, idxbuf, out);
}